// GATNet_12945031430851
// MI455X (gfx1250) — compile-verified
//
#include <hip/hip_runtime.h>
#include <hip/hip_bf16.h>
#include <math.h>

// Problem constants (match reference)
#define NN   32768
#define EE   524288
#define ETOT (EE + NN)   // edges + self-loops = 557056

typedef __attribute__((ext_vector_type(2))) float v2f;
typedef __attribute__((ext_vector_type(8))) float v8f;
typedef __attribute__((ext_vector_type(4))) int   v4i;
typedef __attribute__((address_space(1))) v4i* gv4i_p;   // global 128-bit chunk
typedef __attribute__((address_space(3))) v4i* lv4i_p;   // LDS   128-bit chunk

#define GEMM_BLOCK 128   // 4 wave32

// ---------------------------------------------------------------------------
// float atomic max via integer atomics (works with m initialized to -inf)
// ---------------------------------------------------------------------------
__device__ __forceinline__ void atomic_max_f32(float* addr, float val) {
  if (val >= 0.0f) {
    atomicMax((int*)addr, __float_as_int(val));
  } else {
    atomicMin((unsigned int*)addr, __float_as_uint(val));
  }
}

__global__ void fill_kernel(float* __restrict__ p, float v, int n) {
  int i = blockIdx.x * blockDim.x + threadIdx.x;
  if (i < n) p[i] = v;
}

// W[K,F] -> Wt[F,K] so a B fragment is one contiguous b64 load.
__global__ void transpose_kernel(const float* __restrict__ W,
                                 float* __restrict__ Wt, int K, int F) {
  int idx = blockIdx.x * blockDim.x + threadIdx.x;
  if (idx >= K * F) return;
  int k = idx / F, f = idx - k * F;
  Wt[(size_t)f * K + k] = W[idx];
}

// ---------------------------------------------------------------------------
// FP32 GEMM via V_WMMA_F32_16X16X4_F32.
//  * A strip (16 contiguous rows of X) staged into LDS with
//    GLOBAL_LOAD_ASYNC_TO_LDS_B128 (ASYNCcnt) + s_wait_asynccnt.
//    LDS row stride K+4 -> 16B-aligned async chunks, conflict-free frag reads.
//  * B comes from the transposed weight Wt[F,K]: per 64-wide K chunk, all 16
//    v2f fragments are preloaded into registers in one unrolled clause
//    (one wait for 16 b64 loads), then 16 WMMAs run back-to-back.
//
// Fragment layouts per CDNA5 ISA 7.12.2:
//   A (16x4):  lanes 0-15 -> K=0,1 ; lanes 16-31 -> K=2,3 ; M = lane&15
//   B (4x16):  lanes 0-15 -> K=0,1 ; lanes 16-31 -> K=2,3 ; N = lane&15
//   C/D:       VGPR v -> row v + 8*(lane>=16), col = lane&15
// All branches wave-uniform -> EXEC all 1s at every WMMA (ISA requirement).
// ---------------------------------------------------------------------------
template <int K>
__global__ __launch_bounds__(GEMM_BLOCK) void gemm_wmma_lds(
    const float* __restrict__ X, const float* __restrict__ Wt,
    float* __restrict__ Hm, int F) {
  constexpr int LDS_STRIDE = K + 4;
  __shared__ float Alds[16 * LDS_STRIDE];

  const int mTile = blockIdx.x;
  const int tid   = threadIdx.x;
  const float* src = X + (size_t)mTile * 16 * K;   // 16 contiguous rows

#if defined(__has_builtin) && __has_builtin(__builtin_amdgcn_global_load_async_to_lds_b128)
  {
    constexpr int CPR = K / 4;          // 16B chunks per row
    constexpr int CHUNKS = 16 * CPR;
    for (int c = tid; c < CHUNKS; c += GEMM_BLOCK) {
      const int r = c / CPR;
      const int o = c - r * CPR;        // chunk within row
      __builtin_amdgcn_global_load_async_to_lds_b128(
          (gv4i_p)(src + r * K + o * 4),
          (lv4i_p)(Alds + r * LDS_STRIDE + o * 4),
          0, 0);
    }
    asm volatile("s_wait_asynccnt 0x0" ::: "memory");
  }
#else
  for (int i = tid; i < 16 * K; i += GEMM_BLOCK) {
    const int r = i / K, k = i - r * K;
    Alds[r * LDS_STRIDE + k] = src[i];
  }
#endif
  __syncthreads();

  const int lane  = tid & 31;
  const int wv    = tid >> 5;
  const int half  = lane >> 4;          // 0: K lanes 0/1, 1: K lanes 2/3
  const int l15   = lane & 15;
  const int nTiles = F >> 4;
  const float* arow = Alds + l15 * LDS_STRIDE;

  for (int nTile = wv; nTile < nTiles; nTile += (GEMM_BLOCK / 32)) {
    const int col = (nTile << 4) + l15;
    const float* wtc = Wt + (size_t)col * K + half * 2;   // b64-aligned
    v8f acc = {};
    for (int kc = 0; kc < K; kc += 64) {
      // Batch-preload 16 B fragments (one clause, one wait).
      v2f bf[16];
#pragma unroll
      for (int kk = 0; kk < 16; ++kk)
        bf[kk] = *(const v2f*)(wtc + kc + kk * 4);
      // 16 WMMAs fed from registers + LDS.
#pragma unroll
      for (int kk = 0; kk < 16; ++kk) {
        const int ka = kc + kk * 4 + half * 2;
        v2f a;
        a[0] = arow[ka];
        a[1] = arow[ka + 1];
        // (neg_a, A, neg_b, B, c_mod, C, reuse_a, reuse_b)
        acc = __builtin_amdgcn_wmma_f32_16x16x4_f32(
            false, a, false, bf[kk], (short)0, acc, false, false);
      }
    }
    const int rbase = (mTile << 4) + half * 8;
#pragma unroll
    for (int v = 0; v < 8; ++v)
      Hm[(size_t)(rbase + v) * F + col] = acc[v];
  }
}

// ---------------------------------------------------------------------------
// al_s[n,h] = sum_c h[n,h,c]*a_src[h,c] ; al_d likewise. One thread per (n,h).
// ---------------------------------------------------------------------------
__global__ void attn_coeff_kernel(const float* __restrict__ Hm,
                                  const float* __restrict__ as_,
                                  const float* __restrict__ ad_,
                                  float* __restrict__ als,
                                  float* __restrict__ ald,
                                  int Hh, int C) {
  int idx = blockIdx.x * blockDim.x + threadIdx.x;   // idx = n*Hh + hd
  if (idx >= NN * Hh) return;
  int hd = idx % Hh;
  const float* hp = Hm + (size_t)idx * C;            // h is [N, Hh, C] contiguous
  float s = 0.0f, d = 0.0f;
  for (int c = 0; c < C; ++c) {
    float hv = hp[c];
    s += hv * as_[hd * C + c];
    d += hv * ad_[hd * C + c];
  }
  als[idx] = s;
  ald[idx] = d;
}

__device__ __forceinline__ void edge_endpoints(const int* __restrict__ ei,
                                               int e, int& s, int& d) {
  if (e < EE) { s = ei[e]; d = ei[EE + e]; }
  else        { s = d = e - EE; }          // self-loop
}

// e = leaky_relu(al_s[src]+al_d[dst], 0.2); running segment max into m[dst,h]
__global__ void edge_logits_kernel(const int* __restrict__ ei,
                                   const float* __restrict__ als,
                                   const float* __restrict__ ald,
                                   float* __restrict__ ebuf,
                                   float* __restrict__ m, int Hh) {
  int idx = blockIdx.x * blockDim.x + threadIdx.x;
  if (idx >= ETOT * Hh) return;
  int e = idx / Hh, hd = idx - e * Hh;
  int s, d; edge_endpoints(ei, e, s, d);
  float v = als[s * Hh + hd] + ald[d * Hh + hd];
  v = (v > 0.0f) ? v : 0.2f * v;
  ebuf[idx] = v;
  atomic_max_f32(&m[d * Hh + hd], v);
}

// ex = exp(e - m[dst]); denom[dst] += ex
__global__ void edge_exp_kernel(const int* __restrict__ ei,
                                float* __restrict__ ebuf,
                                const float* __restrict__ m,
                                float* __restrict__ denom, int Hh) {
  int idx = blockIdx.x * blockDim.x + threadIdx.x;
  if (idx >= ETOT * Hh) return;
  int e = idx / Hh, hd = idx - e * Hh;
  int s, d; edge_endpoints(ei, e, s, d);
  float ex = expf(ebuf[idx] - m[d * Hh + hd]);
  ebuf[idx] = ex;
  atomicAdd(&denom[d * Hh + hd], ex);
}

// alpha = ex / denom[dst]
__global__ void edge_alpha_kernel(const int* __restrict__ ei,
                                  float* __restrict__ ebuf,
                                  const float* __restrict__ denom, int Hh) {
  int idx = blockIdx.x * blockDim.x + threadIdx.x;
  if (idx >= ETOT * Hh) return;
  int e = idx / Hh, hd = idx - e * Hh;
  int s, d; edge_endpoints(ei, e, s, d);
  ebuf[idx] = ebuf[idx] / denom[d * Hh + hd];
}

// accum[dst, f..f+3] += alpha[e, f/C] * h[src, f..f+3]
// float4-vectorized: one b128 gather + 4 L2-resident fp32 atomics per thread.
// Channel groups of 4 never straddle a head boundary (C % 4 == 0).
__global__ void scatter_kernel(const int* __restrict__ ei,
                               const float* __restrict__ ebuf,
                               const float* __restrict__ Hm,
                               float* __restrict__ accum, int Hh, int C) {
  const int F  = Hh * C;
  const int Fv = F >> 2;                 // float4 groups per edge
  const int total  = ETOT * Fv;
  const int stride = gridDim.x * blockDim.x;
  for (int i = blockIdx.x * blockDim.x + threadIdx.x; i < total; i += stride) {
    int e = i / Fv;
    int g = i - e * Fv;
    int f = g << 2;
    int hd = f / C;
    int s, d; edge_endpoints(ei, e, s, d);
    float alpha = ebuf[e * Hh + hd];
    const float4 hv = *(const float4*)(Hm + (size_t)s * F + f);
    float* ap = accum + (size_t)d * F + f;
    atomicAdd(ap + 0, alpha * hv.x);
    atomicAdd(ap + 1, alpha * hv.y);
    atomicAdd(ap + 2, alpha * hv.z);
    atomicAdd(ap + 3, alpha * hv.w);
  }
}

// mean over heads + bias + exact-erf GELU (layers 0,1)
__global__ void finalize_mean_gelu_kernel(const float* __restrict__ accum,
                                          const float* __restrict__ b,
                                          float* __restrict__ out,
                                          int Hh, int C) {
  int idx = blockIdx.x * blockDim.x + threadIdx.x;
  if (idx >= NN * C) return;
  int n = idx / C, c = idx - n * C;
  float s = 0.0f;
  for (int h = 0; h < Hh; ++h) s += accum[(size_t)n * Hh * C + h * C + c];
  float x = s / (float)Hh + b[c];
  out[idx] = 0.5f * x * (1.0f + erff(x * 0.70710678118654752f));
}

// concat (H=1) + bias, no activation (layer 2)
__global__ void finalize_bias_kernel(const float* __restrict__ accum,
                                     const float* __restrict__ b,
                                     float* __restrict__ out, int F) {
  int idx = blockIdx.x * blockDim.x + threadIdx.x;
  if (idx >= NN * F) return;
  out[idx] = accum[idx] + b[idx % F];
}

// ---------------------------------------------------------------------------
// Host side
// ---------------------------------------------------------------------------
static void run_layer(const float* x, const float* W, const float* as_,
                      const float* ad_, const float* bias, float* out,
                      int cin, int C, int Hh, bool last,
                      float* h, float* accum, float* als, float* ald,
                      float* m, float* denom, float* ebuf, float* Wt,
                      const int* ei, hipStream_t stream) {
  const int F = Hh * C;
  // 0) transpose weights: Wt[F, cin]
  transpose_kernel<<<(cin * F + 255) / 256, 256, 0, stream>>>(W, Wt, cin, F);
  // 1) h = x @ W  (WMMA fp32, async-LDS A staging). One block per 16-row strip.
  if (cin == 128)
    gemm_wmma_lds<128><<<NN / 16, GEMM_BLOCK, 0, stream>>>(x, Wt, h, F);
  else
    gemm_wmma_lds<64><<<NN / 16, GEMM_BLOCK, 0, stream>>>(x, Wt, h, F);
  // 2) init segment-max / denom / accumulator
  fill_kernel<<<(NN * Hh + 255) / 256, 256, 0, stream>>>(m, -INFINITY, NN * Hh);
  fill_kernel<<<(NN * Hh + 255) / 256, 256, 0, stream>>>(denom, 0.0f, NN * Hh);
  fill_kernel<<<(NN * F + 255) / 256, 256, 0, stream>>>(accum, 0.0f, NN * F);
  // 3) attention coefficients per node/head
  attn_coeff_kernel<<<(NN * Hh + 255) / 256, 256, 0, stream>>>(h, as_, ad_, als, ald, Hh, C);
  // 4) edge softmax
  const int eth = ETOT * Hh;
  edge_logits_kernel<<<(eth + 255) / 256, 256, 0, stream>>>(ei, als, ald, ebuf, m, Hh);
  edge_exp_kernel<<<(eth + 255) / 256, 256, 0, stream>>>(ei, ebuf, m, denom, Hh);
  edge_alpha_kernel<<<(eth + 255) / 256, 256, 0, stream>>>(ei, ebuf, denom, Hh);
  // 5) weighted gather/scatter (float4)
  scatter_kernel<<<4096, 256, 0, stream>>>(ei, ebuf, h, accum, Hh, C);
  // 6) finalize
  if (!last)
    finalize_mean_gelu_kernel<<<(NN * C + 255) / 256, 256, 0, stream>>>(accum, bias, out, Hh, C);
  else
    finalize_bias_kernel<<<(NN * F + 255) / 256, 256, 0, stream>>>(accum, bias, out, F);
}

extern "C" void kernel_launch(void* const* d_in, const int* in_sizes, int n_in,
                              void* d_out, int out_size, void* d_ws, size_t ws_size,
                              hipStream_t stream) {
  (void)in_sizes; (void)n_in; (void)out_size; (void)ws_size;
  const float* x   = (const float*)d_in[0];
  const int*   ei  = (const int*)d_in[1];
  const float* W0  = (const float*)d_in[2];
  const float* as0 = (const float*)d_in[3];
  const float* ad0 = (const float*)d_in[4];
  const float* b0  = (const float*)d_in[5];
  const float* W1  = (const float*)d_in[6];
  const float* as1 = (const float*)d_in[7];
  const float* ad1 = (const float*)d_in[8];
  const float* b1  = (const float*)d_in[9];
  const float* W2  = (const float*)d_in[10];
  const float* as2 = (const float*)d_in[11];
  const float* ad2 = (const float*)d_in[12];
  const float* b2  = (const float*)d_in[13];
  float* out = (float*)d_out;

  // Scratch layout (floats); total ~18.9M floats ~= 75.6 MB
  float* ws    = (float*)d_ws;
  float* h     = ws;                       // N*192
  float* accum = h     + (size_t)NN * 192; // N*192
  float* als   = accum + (size_t)NN * 192; // N*3
  float* ald   = als   + (size_t)NN * 3;   // N*3
  float* m     = ald   + (size_t)NN * 3;   // N*3
  float* denom = m     + (size_t)NN * 3;   // N*3
  float* ebuf  = denom + (size_t)NN * 3;   // Etot*3
  float* xA    = ebuf  + (size_t)ETOT * 3; // N*64
  float* xB    = xA    + (size_t)NN * 64;  // N*64
  float* Wt    = xB    + (size_t)NN * 64;  // 192*128 max

  // Layer 0: cin=128, C=64, H=3, mean+GELU
  run_layer(x,  W0, as0, ad0, b0, xA, 128, 64, 3, false,
            h, accum, als, ald, m, denom, ebuf, Wt, ei, stream);
  // Layer 1: cin=64, C=64, H=3, mean+GELU
  run_layer(xA, W1, as1, ad1, b1, xB, 64, 64, 3, false,
            h, accum, als, ald, m, denom, ebuf, Wt, ei, stream);
  // Layer 2: cin=64, C=64, H=1, concat + bias -> d_out
  run_layer(xB, W2, as2, ad2, b2, out, 64, 64, 1, true,
            h, accum, als, ald, m, denom, ebuf, Wt, ei, stream);
}